// MambaBlock_39659728011562
// MI455X (gfx1250) — compile-verified
//
#include <hip/hip_runtime.h>
#include <cstddef>
#include <cstdint>

#define D_MODEL 1024
#define D_STATE 16
#define D_CONV  4
#define D_INNER 2048
#define BATCH   2
#define SEQ     1024
#define MROWS   (BATCH * SEQ)      // 2048
#define NDBL    (1 + 2 * D_STATE)  // 33

typedef __attribute__((ext_vector_type(2))) float v2f;
typedef __attribute__((ext_vector_type(8))) float v8f;

// ---- CDNA5 async global->LDS DMA (ASYNCcnt-tracked), GV addressing ----
__device__ __forceinline__ void async_load_b128(uint32_t lds_off, const float* gptr) {
  asm volatile("global_load_async_to_lds_b128 %0, %1, off"
               :: "v"(lds_off), "v"(gptr)
               : "memory");
}
__device__ __forceinline__ void wait_asynccnt0() {
  asm volatile("s_wait_asynccnt 0x0" ::: "memory");
}

// =====================================================================
// GEMM (NT):  C[m,n] = sum_k A[m,k] * W[n,k]  (+ resid[m,n] if RESID)
// Block tile 64(M) x 128(N), BK=16, 256 threads = 8 waves.
// Tiles staged by GLOBAL_LOAD_ASYNC_TO_LDS_B128 into double-buffered LDS
// (pitch 20 floats: 16B-aligned B128 targets, conflict-free frag reads).
// Wave (wm,wn) computes a 16x64 strip as 4 WMMA_F32_16X16X4_F32 accums.
// Fragment layout (wave32):
//   A 16x4 : lane<16 -> M=lane,  VGPR r -> K = r      ; lane>=16 -> K = r+2
//   B 4x16 : lane<16 -> N=lane,  VGPR r -> K = r      ; lane>=16 -> K = r+2
//   C 16x16: lane<16 -> N=lane,  VGPR r -> M = r      ; lane>=16 -> M = r+8
// =====================================================================
template <bool RESID>
__global__ __launch_bounds__(256) void gemm_nt_wmma(
    const float* __restrict__ A, const float* __restrict__ W,
    const float* __restrict__ resid, float* __restrict__ C,
    int M, int N, int K)
{
  __shared__ __align__(16) float As[2][64][20];    // [buf][m][k]
  __shared__ __align__(16) float Bt[2][128][20];   // [buf][n][k]

  const int tid  = threadIdx.x;
  const int lane = tid & 31;
  const int wave = tid >> 5;
  const int wm   = wave & 3;      // 0..3  -> M sub-tile
  const int wn   = wave >> 2;     // 0..1  -> N half
  const int half = lane >> 4;     // hi/lo 16 lanes
  const int lq   = lane & 15;

  const int m0 = blockIdx.y * 64;
  const int n0 = blockIdx.x * 128;

  const int ar = tid >> 2;        // 0..63 : row fetched by this thread
  const int ac = (tid & 3) * 4;   // 0,4,8,12 : k offset within tile

  // per-thread global source pointers (advance 16 floats per K-tile)
  const float* aG  = A + (size_t)(m0 + ar) * K + ac;
  const float* bG0 = W + (size_t)(n0 + ar) * K + ac;
  const float* bG1 = W + (size_t)(n0 + ar + 64) * K + ac;

  // per-thread LDS destinations (flat ptr truncation == LDS byte offset)
  uint32_t aL[2], b0L[2], b1L[2];
#pragma unroll
  for (int b = 0; b < 2; ++b) {
    aL[b]  = (uint32_t)(uintptr_t)&As[b][ar][ac];
    b0L[b] = (uint32_t)(uintptr_t)&Bt[b][ar][ac];
    b1L[b] = (uint32_t)(uintptr_t)&Bt[b][ar + 64][ac];
  }

  auto issue = [&](int t) {
    const int buf = t & 1;
    const float* ap  = aG + t * 16;
    const float* bp0 = bG0 + t * 16;
    const float* bp1 = bG1 + t * 16;
    async_load_b128(aL[buf], ap);
    async_load_b128(b0L[buf], bp0);
    async_load_b128(b1L[buf], bp1);
  };

  v8f acc[4] = {};

  const int T = K >> 4;
  issue(0);
  wait_asynccnt0();
  __syncthreads();

  for (int t = 0; t < T; ++t) {
    if (t + 1 < T) {
      issue(t + 1);                 // DMA next tile into other buffer
      if (t + 2 < T) {              // pull tile-after-next toward L2
        __builtin_prefetch(aG + (t + 2) * 16, 0, 1);
        __builtin_prefetch(bG0 + (t + 2) * 16, 0, 1);
        __builtin_prefetch(bG1 + (t + 2) * 16, 0, 1);
      }
    }

    const int cur  = t & 1;
    const int arow = wm * 16 + lq;
#pragma unroll
    for (int kk = 0; kk < 16; kk += 4) {
      const int ka = kk + 2 * half;
      v2f a;
      a.x = As[cur][arow][ka];
      a.y = As[cur][arow][ka + 1];
#pragma unroll
      for (int j = 0; j < 4; ++j) {
        const int bn = wn * 64 + j * 16 + lq;
        v2f b;
        b.x = Bt[cur][bn][ka];
        b.y = Bt[cur][bn][ka + 1];
        acc[j] = __builtin_amdgcn_wmma_f32_16x16x4_f32(
            false, a, false, b, (short)0, acc[j], false, false);
      }
    }

    wait_asynccnt0();   // this wave's DMAs for t+1 have landed in LDS
    __syncthreads();    // all waves: DMAs visible + old buffer fully read
  }

#pragma unroll
  for (int j = 0; j < 4; ++j) {
    const int n = n0 + wn * 64 + j * 16 + lq;
#pragma unroll
    for (int r = 0; r < 8; ++r) {
      const int m = m0 + wm * 16 + r + 8 * half;
      float v = acc[j][r];
      if (RESID) v += resid[(size_t)m * N + n];
      C[(size_t)m * N + n] = v;
    }
  }
}

// =====================================================================
// Depthwise causal conv (D_CONV=4) + bias + SiLU.
// xz rows are 4096 wide; x_inner is the first 2048 channels.
// =====================================================================
__global__ __launch_bounds__(256) void conv_silu_kernel(
    const float* __restrict__ xz, const float* __restrict__ cw,
    const float* __restrict__ cb, float* __restrict__ xc)
{
  const int idx = blockIdx.x * 256 + threadIdx.x;  // over B*L*D_INNER = 2^22
  const int d  = idx & (D_INNER - 1);
  const int l  = (idx >> 11) & (SEQ - 1);
  const int bl = idx >> 11;                        // b*SEQ + l
  const int b  = idx >> 21;

  float s = cb[d];
#pragma unroll
  for (int t = 0; t < D_CONV; ++t) {
    const int ls = l - (D_CONV - 1) + t;
    if (ls >= 0)
      s += xz[((size_t)(b * SEQ + ls) << 12) + d] * cw[d * D_CONV + t];
  }
  xc[((size_t)bl << 11) + d] = s / (1.0f + __expf(-s));
}

// =====================================================================
// x_dbl[m, j] = sum_k xc[m,k] * x_proj_w[j,k]   (N=33, tiny GEMM on VALU)
// =====================================================================
__global__ __launch_bounds__(256) void xproj_kernel(
    const float* __restrict__ xc, const float* __restrict__ w,
    float* __restrict__ xdbl)
{
  const int idx = blockIdx.x * 256 + threadIdx.x;
  if (idx >= MROWS * NDBL) return;
  const int m = idx / NDBL;
  const int j = idx - m * NDBL;
  const float* xr = xc + ((size_t)m << 11);
  const float* wr = w + (size_t)j * D_INNER;
  float s = 0.f;
  for (int k = 0; k < D_INNER; k += 4) {
    s += xr[k] * wr[k] + xr[k + 1] * wr[k + 1] +
         xr[k + 2] * wr[k + 2] + xr[k + 3] * wr[k + 3];
  }
  xdbl[idx] = s;
}

// =====================================================================
// Selective scan: one thread per channel d; 16 states in registers.
// Fuses dt = softplus(x_dbl[:,0]*dtw + dtb), skip D*xc and z-gate SiLU.
// B/C/dt input row (33 floats) staged via double-buffered LDS.
// =====================================================================
__global__ __launch_bounds__(256) void scan_kernel(
    const float* __restrict__ xdbl, const float* __restrict__ xc,
    const float* __restrict__ xz, const float* __restrict__ dtw_,
    const float* __restrict__ dtb_, const float* __restrict__ A_log,
    const float* __restrict__ Dp, float* __restrict__ y)
{
  __shared__ float row[2][NDBL + 1];
  const int tid = threadIdx.x;
  const int b = blockIdx.x >> 3;
  const int d = ((blockIdx.x & 7) << 8) + tid;

  float h[D_STATE];
  float Ad[D_STATE];
#pragma unroll
  for (int n = 0; n < D_STATE; ++n) {
    h[n] = 0.f;
    Ad[n] = -__expf(A_log[d * D_STATE + n]);
  }
  const float dtw = dtw_[d], dtb = dtb_[d], Dd = Dp[d];
  const size_t mbase = (size_t)b * SEQ;

  if (tid < NDBL) row[0][tid] = xdbl[mbase * NDBL + tid];
  __syncthreads();

  for (int l = 0; l < SEQ; ++l) {
    const int cur = l & 1;
    if (l + 1 < SEQ && tid < NDBL)
      row[cur ^ 1][tid] = xdbl[(mbase + l + 1) * NDBL + tid];

    const size_t m = mbase + l;
    const float xin = row[cur][0] * dtw + dtb;
    const float dt = (xin > 20.f) ? xin : log1pf(__expf(xin));
    const float xcv = xc[(m << 11) + d];
    const float dtx = dt * xcv;
    float acc = 0.f;
#pragma unroll
    for (int n = 0; n < D_STATE; ++n) {
      const float dA = __expf(dt * Ad[n]);
      h[n] = dA * h[n] + dtx * row[cur][1 + n];
      acc += h[n] * row[cur][1 + D_STATE + n];
    }
    const float zv = xz[(m << 12) + D_INNER + d];
    const float sz = zv / (1.0f + __expf(-zv));
    y[(m << 11) + d] = (acc + Dd * xcv) * sz;
    __syncthreads();
  }
}

// =====================================================================
// Row LayerNorm over D_MODEL=1024; one block (256 thr) per row.
// =====================================================================
__global__ __launch_bounds__(256) void layernorm_kernel(
    const float* __restrict__ o, const float* __restrict__ g,
    const float* __restrict__ be, float* __restrict__ out)
{
  __shared__ float s1[256], s2[256];
  const int tid = threadIdx.x;
  const size_t r = blockIdx.x;
  const float* orow = o + r * D_MODEL;

  float v[4];
  float sum = 0.f, sq = 0.f;
#pragma unroll
  for (int i = 0; i < 4; ++i) {
    v[i] = orow[tid + i * 256];
    sum += v[i];
    sq += v[i] * v[i];
  }
  s1[tid] = sum;
  s2[tid] = sq;
  __syncthreads();
  for (int off = 128; off > 0; off >>= 1) {
    if (tid < off) { s1[tid] += s1[tid + off]; s2[tid] += s2[tid + off]; }
    __syncthreads();
  }
  const float mu  = s1[0] * (1.0f / D_MODEL);
  const float var = s2[0] * (1.0f / D_MODEL) - mu * mu;
  const float inv = rsqrtf(var + 1e-5f);
#pragma unroll
  for (int i = 0; i < 4; ++i) {
    const int c = tid + i * 256;
    out[r * D_MODEL + c] = (v[i] - mu) * inv * g[c] + be[c];
  }
}

// ===================== workspace layout (floats) =====================
static const size_t XZ_OFF   = 0;                                      // 2048*4096
static const size_t XC_OFF   = XZ_OFF + (size_t)MROWS * 2 * D_INNER;   // 2048*2048
static const size_t XDBL_OFF = XC_OFF + (size_t)MROWS * D_INNER;       // 2048*33
static const size_t Y_OFF    = XDBL_OFF + (size_t)MROWS * NDBL;        // 2048*2048
static const size_t O_OFF    = Y_OFF + (size_t)MROWS * D_INNER;        // 2048*1024

extern "C" void kernel_launch(void* const* d_in, const int* in_sizes, int n_in,
                              void* d_out, int out_size, void* d_ws, size_t ws_size,
                              hipStream_t stream)
{
  const float* x          = (const float*)d_in[0];
  const float* in_proj_w  = (const float*)d_in[1];
  const float* conv_w     = (const float*)d_in[2];
  const float* conv_b     = (const float*)d_in[3];
  const float* x_proj_w   = (const float*)d_in[4];
  const float* dt_proj_w  = (const float*)d_in[5];
  const float* dt_proj_b  = (const float*)d_in[6];
  const float* A_log      = (const float*)d_in[7];
  const float* Dp         = (const float*)d_in[8];
  const float* out_proj_w = (const float*)d_in[9];
  const float* ln_gamma   = (const float*)d_in[10];
  const float* ln_beta    = (const float*)d_in[11];

  float* ws   = (float*)d_ws;
  float* xz   = ws + XZ_OFF;
  float* xc   = ws + XC_OFF;
  float* xdbl = ws + XDBL_OFF;
  float* ybuf = ws + Y_OFF;
  float* obuf = ws + O_OFF;

  // 1) xz = x @ in_proj_w.T        (2048x1024 * 1024x4096)
  gemm_nt_wmma<false><<<dim3(4096 / 128, 2048 / 64), 256, 0, stream>>>(
      x, in_proj_w, nullptr, xz, MROWS, 2 * D_INNER, D_MODEL);

  // 2) causal depthwise conv + bias + SiLU -> xc
  conv_silu_kernel<<<(MROWS * D_INNER) / 256, 256, 0, stream>>>(
      xz, conv_w, conv_b, xc);

  // 3) x_dbl = xc @ x_proj_w.T     (N=33)
  xproj_kernel<<<(MROWS * NDBL + 255) / 256, 256, 0, stream>>>(
      xc, x_proj_w, xdbl);

  // 4) selective scan (fuses dt softplus, D-skip, z-gate) -> y
  scan_kernel<<<BATCH * (D_INNER / 256), 256, 0, stream>>>(
      xdbl, xc, xz, dt_proj_w, dt_proj_b, A_log, Dp, ybuf);

  // 5) o = y @ out_proj_w.T + residual x
  gemm_nt_wmma<true><<<dim3(1024 / 128, 2048 / 64), 256, 0, stream>>>(
      ybuf, out_proj_w, x, obuf, MROWS, D_MODEL, D_INNER);

  // 6) LayerNorm -> d_out
  layernorm_kernel<<<MROWS, 256, 0, stream>>>(
      obuf, ln_gamma, ln_beta, (float*)d_out);
}